// TreeLSTMCell_42485816492070
// MI455X (gfx1250) — compile-verified
//
#include <hip/hip_runtime.h>

typedef float v2f __attribute__((ext_vector_type(2)));
typedef float v8f __attribute__((ext_vector_type(8)));

__device__ __forceinline__ v8f wmma_f32_k4(v2f a, v2f b, v8f c) {
  // D = A(16x4 f32) x B(4x16 f32) + C(16x16 f32)
  return __builtin_amdgcn_wmma_f32_16x16x4_f32(false, a, false, b, (short)0, c, false, false);
}

__device__ __forceinline__ float fast_tanhf(float x) {
#if __has_builtin(__builtin_amdgcn_tanhf)
  return __builtin_amdgcn_tanhf(x);           // v_tanh_f32 (gfx1250 TRANS op)
#else
  float e = __expf(-2.0f * x);
  return (1.0f - e) / (1.0f + e);
#endif
}

__device__ __forceinline__ float sigmoidf_(float x) {
#if __has_builtin(__builtin_amdgcn_tanhf)
  return __builtin_fmaf(0.5f, __builtin_amdgcn_tanhf(0.5f * x), 0.5f);
#else
  return 1.0f / (1.0f + __expf(-x));
#endif
}

// A-fragment pair (row r = lane&15, features f0..f0+1). FULL: no row guard.
template <bool FULL>
__device__ __forceinline__ v2f load_a_pair(const float* __restrict__ base, int r, int N,
                                           int F, int f0) {
  v2f z = {0.0f, 0.0f};
  int f0c = (f0 < F) ? f0 : 0;              // pairs fully valid or fully invalid
  if (FULL) {
    v2f v = *(const v2f*)(base + (unsigned)(r * F + f0c));
    return (f0 < F) ? v : z;                // const-folds unless F==10, s==2, half==1
  } else {
    int rc = (r < N) ? r : (N - 1);
    v2f v = *(const v2f*)(base + (unsigned)(rc * F + f0c));
    return ((r < N) && (f0 < F)) ? v : z;
  }
}

// B-fragment pair: B[f][g] = W[g][f]; W is [G,F] row-major. Lane-mask only.
__device__ __forceinline__ v2f load_b_pair(const float* __restrict__ W, int G, int F,
                                           int f0, int g) {
  int gc  = (g < G) ? g : (G - 1);
  int f0c = (f0 < F) ? f0 : 0;
  v2f v = *(const v2f*)(W + (unsigned)(gc * F + f0c));
  v2f z = {0.0f, 0.0f};
  return ((g < G) && (f0 < F)) ? v : z;
}

// c_children element in C/D layout position (row r, col g); col range is [0,20).
template <bool FULL>
__device__ __forceinline__ float load_c_elem(const float* __restrict__ ck, int r, int N, int g) {
  int gc = (g < 20) ? g : 19;
  if (FULL) {
    float v = ck[(unsigned)(r * 20 + gc)];
    return (g < 20) ? v : 0.0f;             // folds for t==0 tile
  } else {
    int rc = (r < N) ? r : (N - 1);
    float v = ck[(unsigned)(rc * 20 + gc)];
    return ((r < N) && (g < 20)) ? v : 0.0f;
  }
}

#define WAVES_PER_BLOCK 8
#define TILE_M 16

template <bool FULL>
__device__ __forceinline__ void tile_body(
    const float* __restrict__ x,  const float* __restrict__ hch, const float* __restrict__ cch,
    const float* __restrict__ Wiou_w, const float* __restrict__ Wiou_b,
    const float* __restrict__ Uiou_w, const float* __restrict__ Uiou_b,
    const float* __restrict__ Wf_w,   const float* __restrict__ Wf_b,
    const float* __restrict__ Uf_w,   const float* __restrict__ Uf_b,
    float* __restrict__ out_h, float* __restrict__ out_c,
    float* __restrict__ liou,  float* __restrict__ lcc,   // this wave's LDS: [16][64], [16][32]
    int node_base, int lane, int N)
{
  const int half_ = lane >> 4;     // 0: K pair {0,1}, 1: K pair {2,3}
  const int nlo   = lane & 15;     // A row / B-C-D column within tile
  const int f0A   = half_ * 2;
  const int rA    = node_base + nlo;

  v8f zero8 = {0.f,0.f,0.f,0.f,0.f,0.f,0.f,0.f};

  // ---- x A-fragments (IN=10 -> kb = 0,4,8; cols 10,11 zero-masked)
  v2f xa[3];
  #pragma unroll
  for (int s = 0; s < 3; ++s)
    xa[s] = load_a_pair<FULL>(x, rA, N, 10, s*4 + f0A);

  // ---- fx = x @ Wf^T : [16x20] over 2 N-tiles
  v8f fx[2] = {zero8, zero8};
  #pragma unroll
  for (int t = 0; t < 2; ++t)
    #pragma unroll
    for (int s = 0; s < 3; ++s)
      fx[t] = wmma_f32_k4(xa[s], load_b_pair(Wf_w, 20, 10, s*4 + f0A, t*16 + nlo), fx[t]);

  // ---- iou = x @ Wiou^T : [16x60] over 4 N-tiles
  v8f iou[4] = {zero8, zero8, zero8, zero8};
  #pragma unroll
  for (int t = 0; t < 4; ++t)
    #pragma unroll
    for (int s = 0; s < 3; ++s)
      iou[t] = wmma_f32_k4(xa[s], load_b_pair(Wiou_w, 60, 10, s*4 + f0A, t*16 + nlo), iou[t]);

  // ---- preload Uf B-fragments (K=20 -> 5 steps, 2 N-tiles), reused for all 4 children
  v2f ufb[5][2];
  #pragma unroll
  for (int s = 0; s < 5; ++s)
    #pragma unroll
    for (int t = 0; t < 2; ++t)
      ufb[s][t] = load_b_pair(Uf_w, 20, 20, s*4 + f0A, t*16 + nlo);

  // per-column forget-gate bias (Wf_b + Uf_b); constant across the 16 rows of a tile
  float fbias[2];
  #pragma unroll
  for (int t = 0; t < 2; ++t) {
    int g  = t*16 + nlo;
    int gc = (g < 20) ? g : 19;
    float b = Wf_b[gc] + Uf_b[gc];
    fbias[t] = (g < 20) ? b : 0.f;
  }

  v2f hs[5];
  #pragma unroll
  for (int s = 0; s < 5; ++s) { hs[s].x = 0.f; hs[s].y = 0.f; }
  v8f cacc[2] = {zero8, zero8};

  // ---- per-child: f_k = sigmoid(fx + h_k @ Uf^T + b), c_acc += f_k * c_k
  #pragma unroll
  for (int k = 0; k < 4; ++k) {
    const float* hk = hch + (unsigned)k * (unsigned)N * 20u;
    const float* ck = cch + (unsigned)k * (unsigned)N * 20u;

    v2f ha[5];
    #pragma unroll
    for (int s = 0; s < 5; ++s) {
      ha[s] = load_a_pair<FULL>(hk, rA, N, 20, s*4 + f0A);
      hs[s] += ha[s];
    }

    v8f acc[2] = {fx[0], fx[1]};
    #pragma unroll
    for (int t = 0; t < 2; ++t)
      #pragma unroll
      for (int s = 0; s < 5; ++s)
        acc[t] = wmma_f32_k4(ha[s], ufb[s][t], acc[t]);

    #pragma unroll
    for (int t = 0; t < 2; ++t) {
      int g = t*16 + nlo;
      #pragma unroll
      for (int v = 0; v < 8; ++v) {
        int r = node_base + v + half_*8;            // C/D layout: M = v + 8*half
        float cv = load_c_elem<FULL>(ck, r, N, g);
        float fg = sigmoidf_(acc[t][v] + fbias[t]);
        cacc[t][v] += fg * cv;
      }
    }
  }

  // ---- iou += h_sum @ Uiou^T
  #pragma unroll
  for (int t = 0; t < 4; ++t)
    #pragma unroll
    for (int s = 0; s < 5; ++s)
      iou[t] = wmma_f32_k4(hs[s], load_b_pair(Uiou_w, 60, 20, s*4 + f0A, t*16 + nlo), iou[t]);

  // ---- spill accumulators to LDS (i/o/u split at 20/40 crosses tile+lane boundaries)
  #pragma unroll
  for (int t = 0; t < 4; ++t) {
    int g  = t*16 + nlo;
    int gc = (g < 60) ? g : 59;
    float b = Wiou_b[gc] + 4.0f * Uiou_b[gc];
    float bias = (g < 60) ? b : 0.f;
    #pragma unroll
    for (int v = 0; v < 8; ++v)
      liou[(v + half_*8) * 64 + g] = iou[t][v] + bias;
  }
  #pragma unroll
  for (int t = 0; t < 2; ++t) {
    int g = t*16 + nlo;
    #pragma unroll
    for (int v = 0; v < 8; ++v)
      lcc[(v + half_*8) * 32 + g] = cacc[t][v];
  }
  // Same-wave LDS traffic is in-order; compiler inserts dscnt waits for the RAW.

  // ---- combine + store in (g,g+1) pairs: 16 rows x 10 pairs = 160 -> 5 per lane.
  // All pairs are 8B-aligned (even g; LDS rows 64/32 floats; out rows 20 floats).
  #pragma unroll
  for (int it = 0; it < 5; ++it) {
    int idx = it * 32 + lane;       // 0..159
    int m   = idx / 10;
    int g   = (idx - m * 10) * 2;
    v2f iv = *(const v2f*)&liou[m * 64 + g];
    v2f ov = *(const v2f*)&liou[m * 64 + 20 + g];
    v2f uv = *(const v2f*)&liou[m * 64 + 40 + g];
    v2f cs = *(const v2f*)&lcc [m * 32 + g];
    v2f c, h;
    c.x = sigmoidf_(iv.x) * fast_tanhf(uv.x) + cs.x;
    c.y = sigmoidf_(iv.y) * fast_tanhf(uv.y) + cs.y;
    h.x = sigmoidf_(ov.x) * fast_tanhf(c.x);
    h.y = sigmoidf_(ov.y) * fast_tanhf(c.y);
    int r = node_base + m;
    unsigned o = (unsigned)(r * 20 + g);
    if (FULL || r < N) {
      *(v2f*)(out_h + o) = h;
      *(v2f*)(out_c + o) = c;
    }
  }
}

__global__ void __launch_bounds__(256) treelstm_wmma_kernel(
    const float* __restrict__ x,        // [N,10]
    const float* __restrict__ hch,      // [4,N,20]
    const float* __restrict__ cch,      // [4,N,20]
    const float* __restrict__ Wiou_w,   // [60,10]
    const float* __restrict__ Wiou_b,   // [60]
    const float* __restrict__ Uiou_w,   // [60,20]
    const float* __restrict__ Uiou_b,   // [60]
    const float* __restrict__ Wf_w,     // [20,10]
    const float* __restrict__ Wf_b,     // [20]
    const float* __restrict__ Uf_w,     // [20,20]
    const float* __restrict__ Uf_b,     // [20]
    float* __restrict__ out_h,          // [N,20]
    float* __restrict__ out_c,          // [N,20]
    int N)
{
  __shared__ float lds_iou[WAVES_PER_BLOCK][16][64];  // 32 KB
  __shared__ float lds_cc [WAVES_PER_BLOCK][16][32];  // 16 KB

  const int lane = threadIdx.x & 31;
  const int wave = threadIdx.x >> 5;
  const int node_base = ((int)blockIdx.x * WAVES_PER_BLOCK + wave) * TILE_M;

  if (node_base >= N) return;                        // wave-uniform; no block barrier used

  float* liou = &lds_iou[wave][0][0];
  float* lcc  = &lds_cc [wave][0][0];

  if (node_base + TILE_M <= N) {                     // hot path: no row guards at all
    tile_body<true >(x, hch, cch, Wiou_w, Wiou_b, Uiou_w, Uiou_b,
                     Wf_w, Wf_b, Uf_w, Uf_b, out_h, out_c, liou, lcc,
                     node_base, lane, N);
  } else {                                           // cold tail: clamped + masked
    tile_body<false>(x, hch, cch, Wiou_w, Wiou_b, Uiou_w, Uiou_b,
                     Wf_w, Wf_b, Uf_w, Uf_b, out_h, out_c, liou, lcc,
                     node_base, lane, N);
  }
}

extern "C" void kernel_launch(void* const* d_in, const int* in_sizes, int n_in,
                              void* d_out, int out_size, void* d_ws, size_t ws_size,
                              hipStream_t stream) {
  const float* x      = (const float*)d_in[0];
  const float* hch    = (const float*)d_in[1];
  const float* cch    = (const float*)d_in[2];
  const float* Wiou_w = (const float*)d_in[3];
  const float* Wiou_b = (const float*)d_in[4];
  const float* Uiou_w = (const float*)d_in[5];
  const float* Uiou_b = (const float*)d_in[6];
  const float* Wf_w   = (const float*)d_in[7];
  const float* Wf_b   = (const float*)d_in[8];
  const float* Uf_w   = (const float*)d_in[9];
  const float* Uf_b   = (const float*)d_in[10];

  const int N = in_sizes[0] / 10;
  float* out_h = (float*)d_out;
  float* out_c = out_h + (size_t)N * 20;

  const int nodes_per_block = WAVES_PER_BLOCK * TILE_M;  // 128
  const int grid = (N + nodes_per_block - 1) / nodes_per_block;

  hipLaunchKernelGGL(treelstm_wmma_kernel, dim3(grid), dim3(256), 0, stream,
                     x, hch, cch, Wiou_w, Wiou_b, Uiou_w, Uiou_b,
                     Wf_w, Wf_b, Uf_w, Uf_b, out_h, out_c, N);
}